// DenseFlashAttention_37580963840116
// MI455X (gfx1250) — compile-verified
//
#include <hip/hip_runtime.h>
#include <math.h>

// ---------------------------------------------------------------------------
// GAT-style graph attention for MI455X (gfx1250, wave32).
//  - fp32 WMMA (v_wmma_f32_16x16x4_f32) for dense GEMM phases; A tiles staged
//    through LDS with coalesced b128 loads.
//  - wave-per-edge gather/scatter; segment softmax uses native CDNA5 float
//    atomics via inline asm (global_atomic_max_num_f32 / global_atomic_add_f32,
//    scope:SCOPE_DEV, no-return -> STOREcnt only).
// ---------------------------------------------------------------------------

#define N_NODES 50000
#define N_EDGES 800000
#define FEAT    32
#define HEADS   4
#define MTILES  (N_NODES / 16)   // 3125 exact

typedef __attribute__((ext_vector_type(2))) float v2f;
typedef __attribute__((ext_vector_type(8))) float v8f;

// Native CDNA5 no-return float atomics (device scope, performed at L2).
__device__ __forceinline__ void atomic_add_f32_dev(float* p, float v) {
  asm volatile("global_atomic_add_f32 %0, %1, off scope:SCOPE_DEV"
               :: "v"(p), "v"(v) : "memory");
}
__device__ __forceinline__ void atomic_max_f32_dev(float* p, float v) {
  asm volatile("global_atomic_max_num_f32 %0, %1, off scope:SCOPE_DEV"
               :: "v"(p), "v"(v) : "memory");
}

__device__ __forceinline__ v8f wmma_f32_k32(const float* Arow,              // A 16xFEAT tile (LDS), stride FEAT
                                            const float* __restrict__ B,   // B FEATxFEAT, row-major
                                            int bcol, int lane, v8f acc) {
  // f32 16x16x4 layout: lane l -> A row (l&15), K pair 2*(l>>4); B mirrors on K.
  const int row = lane & 15;
  const int hi  = lane >> 4;
#pragma unroll
  for (int kc = 0; kc < 8; ++kc) {
    const int k0 = kc * 4 + 2 * hi;
    v2f a, b;
    a.x = Arow[row * FEAT + k0];
    a.y = Arow[row * FEAT + k0 + 1];
    b.x = B[k0 * FEAT + bcol + row];        // "row" doubles as the N column here
    b.y = B[(k0 + 1) * FEAT + bcol + row];
    acc = __builtin_amdgcn_wmma_f32_16x16x4_f32(false, a, false, b, (short)0, acc,
                                                false, false);
  }
  return acc;
}

// --- 0: re-init atomic targets each call (graph replay safe) ----------------
__global__ void gat_init(float* __restrict__ m, float* __restrict__ denom,
                         float* __restrict__ num) {
  const int idx = blockIdx.x * blockDim.x + threadIdx.x;
  if (idx < N_NODES * HEADS) {
    m[idx]     = -INFINITY;
    denom[idx] = 0.0f;
  }
  if (idx < N_NODES * HEADS * FEAT) num[idx] = 0.0f;
}

// --- 1: Q/K/V projections via f32 WMMA --------------------------------------
// qkv layout: [mat(0=Q,1=K,2=V)][head][node][feat]
__global__ __launch_bounds__(128) void gat_qkv_wmma(const float* __restrict__ x,
                                                    const float* __restrict__ Wq,
                                                    const float* __restrict__ Wk,
                                                    const float* __restrict__ Wv,
                                                    float* __restrict__ qkv) {
  const int lane = threadIdx.x & 31;
  const int wave = threadIdx.x >> 5;           // 0..3
  const int comb = blockIdx.y * 4 + wave;      // 0..23 = mat*8 + head*2 + nt
  const int mat  = comb >> 3;
  const int head = (comb >> 1) & 3;
  const int nt   = comb & 1;
  const int m0   = blockIdx.x * 16;

  // Stage the contiguous 16x32 fp32 x-tile (2 KB) through LDS, coalesced b128.
  __shared__ float xs[16 * FEAT];
  ((float4*)xs)[threadIdx.x] = ((const float4*)(x + (size_t)m0 * FEAT))[threadIdx.x];
  __syncthreads();

  const float* W = (mat == 0 ? Wq : (mat == 1 ? Wk : Wv)) + head * FEAT * FEAT;
  v8f acc = {};
  acc = wmma_f32_k32(xs, W, nt * 16, lane, acc);

  float* out = qkv + ((size_t)(mat * HEADS + head) * N_NODES) * FEAT;
  const int col = lane & 15, hi = lane >> 4;
#pragma unroll
  for (int r = 0; r < 8; ++r)
    out[(size_t)(m0 + r + 8 * hi) * FEAT + nt * 16 + col] = acc[r];
}

// --- 2: per-edge logits + per-receiver running max --------------------------
// one wave per edge; lane -> (head = l>>3, feature quad = l&7)
__global__ __launch_bounds__(256) void gat_edge_logits(const float* __restrict__ qkv,
                                                       const int* __restrict__ ei,
                                                       float* __restrict__ sEH,
                                                       float* __restrict__ m) {
  const int wid = (blockIdx.x * blockDim.x + threadIdx.x) >> 5;
  if (wid >= N_EDGES) return;
  const int lane = threadIdx.x & 31;
  if (lane == 0 && wid + 8192 < N_EDGES) {      // stream the index arrays ahead
    __builtin_prefetch(ei + wid + 8192, 0, 0);
    __builtin_prefetch(ei + N_EDGES + wid + 8192, 0, 0);
  }
  const int h    = lane >> 3;
  const int f4   = (lane & 7) * 4;
  const int snd  = ei[wid];
  const int rcv  = ei[N_EDGES + wid];

  const float* Qr = qkv + ((size_t)(0 * HEADS + h) * N_NODES + rcv) * FEAT;
  const float* Kr = qkv + ((size_t)(1 * HEADS + h) * N_NODES + snd) * FEAT;
  const float4 q = *(const float4*)(Qr + f4);
  const float4 k = *(const float4*)(Kr + f4);
  float p = q.x * k.x + q.y * k.y + q.z * k.z + q.w * k.w;
  p += __shfl_xor(p, 1, 32);
  p += __shfl_xor(p, 2, 32);
  p += __shfl_xor(p, 4, 32);
  if ((lane & 7) == 0) {
    const float s = p * 0.17677669529663687f;  // 32^-0.5
    sEH[(size_t)wid * HEADS + h] = s;
    atomic_max_f32_dev(&m[rcv * HEADS + h], s);
  }
}

// --- 3: exp-weighted scatter of V into (num, denom) -------------------------
__global__ __launch_bounds__(256) void gat_edge_scatter(const float* __restrict__ qkv,
                                                        const int* __restrict__ ei,
                                                        const float* __restrict__ sEH,
                                                        const float* __restrict__ m,
                                                        float* __restrict__ denom,
                                                        float* __restrict__ num) {
  const int wid = (blockIdx.x * blockDim.x + threadIdx.x) >> 5;
  if (wid >= N_EDGES) return;
  const int lane = threadIdx.x & 31;
  if (lane == 0 && wid + 8192 < N_EDGES) {
    __builtin_prefetch(ei + wid + 8192, 0, 0);
    __builtin_prefetch(ei + N_EDGES + wid + 8192, 0, 0);
    __builtin_prefetch(sEH + (size_t)(wid + 8192) * HEADS, 0, 0);
  }
  const int h    = lane >> 3;
  const int f4   = (lane & 7) * 4;
  const int snd  = ei[wid];
  const int rcv  = ei[N_EDGES + wid];

  const float s  = sEH[(size_t)wid * HEADS + h];
  const float mv = m[rcv * HEADS + h];
  const float w  = __expf(s - mv);
  if ((lane & 7) == 0) atomic_add_f32_dev(&denom[rcv * HEADS + h], w);

  const float* Vr = qkv + ((size_t)(2 * HEADS + h) * N_NODES + snd) * FEAT;
  const float4 v  = *(const float4*)(Vr + f4);
  float* dst = num + ((size_t)rcv * HEADS + h) * FEAT + f4;
  atomic_add_f32_dev(dst + 0, w * v.x);
  atomic_add_f32_dev(dst + 1, w * v.y);
  atomic_add_f32_dev(dst + 2, w * v.z);
  atomic_add_f32_dev(dst + 3, w * v.w);
}

// --- 4: normalize + mean over heads -----------------------------------------
__global__ void gat_finalize(const float* __restrict__ num,
                             const float* __restrict__ denom,
                             float* __restrict__ attnMean) {
  const int idx = blockIdx.x * blockDim.x + threadIdx.x;
  if (idx >= N_NODES * FEAT) return;
  const int n = idx >> 5, f = idx & 31;
  float acc = 0.0f;
#pragma unroll
  for (int h = 0; h < HEADS; ++h) {
    const float d = denom[n * HEADS + h];
    const float v = num[((size_t)n * HEADS + h) * FEAT + f];
    acc += (d > 0.0f) ? v / d : 0.0f;
  }
  attnMean[idx] = acc * 0.25f;
}

// --- 5: out = x + attnMean @ Wout via f32 WMMA ------------------------------
__global__ __launch_bounds__(64) void gat_out_wmma(const float* __restrict__ x,
                                                   const float* __restrict__ attnMean,
                                                   const float* __restrict__ Wout,
                                                   float* __restrict__ out) {
  const int lane = threadIdx.x & 31;
  const int nt   = threadIdx.x >> 5;  // 0..1
  const int m0   = blockIdx.x * 16;

  __shared__ float as[16 * FEAT];
  {
    const float4* src = (const float4*)(attnMean + (size_t)m0 * FEAT);
    ((float4*)as)[threadIdx.x]      = src[threadIdx.x];
    ((float4*)as)[threadIdx.x + 64] = src[threadIdx.x + 64];
  }
  __syncthreads();

  v8f acc = {};
  acc = wmma_f32_k32(as, Wout, nt * 16, lane, acc);

  const int col = lane & 15, hi = lane >> 4;
#pragma unroll
  for (int r = 0; r < 8; ++r) {
    const size_t o = (size_t)(m0 + r + 8 * hi) * FEAT + nt * 16 + col;
    out[o] = x[o] + acc[r];
  }
}

// ---------------------------------------------------------------------------
extern "C" void kernel_launch(void* const* d_in, const int* in_sizes, int n_in,
                              void* d_out, int out_size, void* d_ws, size_t ws_size,
                              hipStream_t stream) {
  const float* x    = (const float*)d_in[0];
  const int*   ei   = (const int*)d_in[1];
  const float* Wq   = (const float*)d_in[2];
  const float* Wk   = (const float*)d_in[3];
  const float* Wv   = (const float*)d_in[4];
  const float* Wout = (const float*)d_in[5];
  float*       out  = (float*)d_out;

  // workspace layout (floats)
  float* ws   = (float*)d_ws;
  float* qkv  = ws;                                        // 3*H*N*F = 19.2M
  float* sEH  = qkv + (size_t)3 * HEADS * N_NODES * FEAT;  // E*H      = 3.2M
  float* mbuf = sEH + (size_t)N_EDGES * HEADS;             // N*H
  float* den  = mbuf + (size_t)N_NODES * HEADS;            // N*H
  float* num  = den + (size_t)N_NODES * HEADS;             // N*H*F    = 6.4M
  float* attn = num + (size_t)N_NODES * HEADS * FEAT;      // N*F      = 1.6M

  gat_init<<<(N_NODES * HEADS * FEAT + 255) / 256, 256, 0, stream>>>(mbuf, den, num);

  gat_qkv_wmma<<<dim3(MTILES, 6), 128, 0, stream>>>(x, Wq, Wk, Wv, qkv);

  const int edgeBlocks = (N_EDGES * 32) / 256;  // one wave per edge
  gat_edge_logits<<<edgeBlocks, 256, 0, stream>>>(qkv, ei, sEH, mbuf);
  gat_edge_scatter<<<edgeBlocks, 256, 0, stream>>>(qkv, ei, sEH, mbuf, den, num);

  gat_finalize<<<(N_NODES * FEAT + 255) / 256, 256, 0, stream>>>(num, den, attn);

  gat_out_wmma<<<MTILES, 64, 0, stream>>>(x, attn, Wout, out);
}